// SO3PredictionModel_91276644974780
// MI455X (gfx1250) — compile-verified
//
#include <hip/hip_runtime.h>
#include <hip/hip_bf16.h>

#define B_   16
#define T_   2048
#define D_   384
#define NB_  128
#define CPW  1152      // packed columns: biv(384) | dec(384) | inj(384)
#define EP   1156      // sE row pitch (pad: 1156 % 64 == 4 -> conflict-free stores)
#define HP   388       // sH row pitch (pad: 388 % 64 == 4 -> conflict-free reads)
#define CH   16        // timesteps per chunk (== WMMA M)
#define EPSF 1e-8f

typedef __attribute__((ext_vector_type(2))) float v2f;
typedef __attribute__((ext_vector_type(8))) float v8f;

// ---------------------------------------------------------------------------
// Kernel 1: fold W_in into the three downstream projections (linear chain:
// no nonlinearity between them, so the 3 big GEMMs collapse to K=4).
//   weff[k][c] = sum_d W_in[k][d] * Wcat[d][c],  Wcat = [W_biv | W_dec | W_inj]
//   bias[c]    = [b_biv | b_dec | b_inj]
// ---------------------------------------------------------------------------
__global__ __launch_bounds__(256) void so3_prep_kernel(
    const float* __restrict__ W_in,
    const float* __restrict__ W_biv, const float* __restrict__ b_biv,
    const float* __restrict__ W_dec, const float* __restrict__ b_dec,
    const float* __restrict__ W_inj, const float* __restrict__ b_inj,
    float* __restrict__ weff, float* __restrict__ bias) {
  int idx = blockIdx.x * blockDim.x + threadIdx.x;
  if (idx >= 4 * CPW) return;
  int k = idx / CPW;
  int c = idx % CPW;
  const float* Wc;
  int cc;
  if (c < D_)            { Wc = W_biv; cc = c; }
  else if (c < 2 * D_)   { Wc = W_dec; cc = c - D_; }
  else                   { Wc = W_inj; cc = c - 2 * D_; }
  float s = 0.f;
  for (int d = 0; d < D_; ++d)
    s = fmaf(W_in[k * D_ + d], Wc[d * D_ + cc], s);
  weff[k * CPW + c] = s;
  if (k == 0)
    bias[c] = (c < D_) ? b_biv[c] : (c < 2 * D_ ? b_dec[c - D_] : b_inj[c - 2 * D_]);
}

// ---------------------------------------------------------------------------
// Kernel 2: fused generate + scan + output projection. One workgroup per
// batch element (16 WGs, 128 threads = 4 wave32). Everything stays in
// LDS/registers; only quat is read and the normalized output written.
// ---------------------------------------------------------------------------
__global__ __launch_bounds__(128) void so3_scan_kernel(
    const float* __restrict__ quat,   // (B,T,4)
    const float* __restrict__ weff,   // (4,1152)
    const float* __restrict__ bias,   // (1152)
    const float* __restrict__ W_out,  // (384,4)
    float* __restrict__ out) {        // (B,T,4)
  __shared__ float sWeff[4 * CPW];    // 18.4 KB
  __shared__ float sBias[CPW];        //  4.6 KB
  __shared__ float sE[CH * EP];       // 74.0 KB  E = quat @ weff + bias, one chunk
  __shared__ float sH[CH * HP];       // 24.8 KB  hidden states of the chunk
  __shared__ float sWout[D_ * 4];     //  6.1 KB
  __shared__ float sPart[4 * 256];    //  4.1 KB  per-wave projection partials

  const int tid   = threadIdx.x;
  const int lane  = tid & 31;                              // wave32
  const int uwave = __builtin_amdgcn_readfirstlane(tid >> 5);  // SGPR wave id 0..3
  const int t16   = lane & 15;        // row / col index inside 16x16 tile
  const int k0    = (lane < 16) ? 0 : 2;   // A/B VGPR K split per half-wave
  const int mhi   = (lane < 16) ? 0 : 8;   // C/D row split per half-wave
  const int b     = blockIdx.x;

  for (int i = tid; i < 4 * CPW; i += 128) sWeff[i] = weff[i];
  for (int i = tid; i < CPW;     i += 128) sBias[i] = bias[i];
  for (int i = tid; i < D_ * 4;  i += 128) sWout[i] = W_out[i];
  __syncthreads();

  float hx = 0.f, hy = 0.f, hz = 0.f;     // thread tid owns 3-block tid
  const float* qbase = quat + (size_t)b * T_ * 4;

  for (int ch = 0; ch < T_ / CH; ++ch) {
    const int t0 = ch * CH;

    // A operand: 16 timesteps x 4 quat components (f32 WMMA A layout:
    // lanes 0-15 hold K=0,1; lanes 16-31 hold K=2,3 for M=lane%16)
    v2f a;
    a.x = qbase[(t0 + t16) * 4 + k0];
    a.y = qbase[(t0 + t16) * 4 + k0 + 1];

    if (tid == 0 && ch + 1 < T_ / CH)
      __builtin_prefetch(qbase + (t0 + CH) * 4, 0, 1);   // global_prefetch_b8

    // E(16 x 1152) = Q(16x4) @ Weff(4x1152) + bias; K=4 f32 WMMA per 16-col
    // tile. uwave is an SGPR -> uniform scalar-branch loop, EXEC stays full.
    for (int tile = uwave; tile < CPW / 16; tile += 4) {
      int col = tile * 16 + t16;
      v2f bm;
      bm.x = sWeff[k0 * CPW + col];
      bm.y = sWeff[(k0 + 1) * CPW + col];
      float bv = sBias[col];
      v8f c = {bv, bv, bv, bv, bv, bv, bv, bv};   // bias broadcast over rows
      v8f d = __builtin_amdgcn_wmma_f32_16x16x4_f32(
          /*neg_a=*/false, a, /*neg_b=*/false, bm,
          /*c_mod=*/(short)0, c, /*reuse_a=*/false, /*reuse_b=*/false);
#pragma unroll
      for (int v = 0; v < 8; ++v)
        sE[(v + mhi) * EP + col] = d[v];
    }
    __syncthreads();

    // Sequential recurrence over the 16 timesteps of this chunk.
    // Blocks are independent -> no barriers inside the step loop.
    const int j3 = tid * 3;
#pragma unroll 4
    for (int tt = 0; tt < CH; ++tt) {
      const float* e = &sE[tt * EP];
      float bx = e[j3], by = e[j3 + 1], bz = e[j3 + 2];
      float nb = sqrtf(bx * bx + by * by + bz * bz);
      nb = fmaxf(nb, EPSF);
      float sn, cs;
      __sincosf(0.5f * nb, &sn, &cs);
      float s  = sn / nb;
      float qx = s * bz, qy = -s * by, qz = s * bx;
      float tx = 2.f * (qy * hz - qz * hy);
      float ty = 2.f * (qz * hx - qx * hz);
      float tz = 2.f * (qx * hy - qy * hx);
      float rx = hx + cs * tx + (qy * tz - qz * ty);
      float ry = hy + cs * ty + (qz * tx - qx * tz);
      float rz = hz + cs * tz + (qx * ty - qy * tx);
      float dx = e[D_ + j3], dy = e[D_ + j3 + 1], dz = e[D_ + j3 + 2];
      float ix = e[2 * D_ + j3], iy = e[2 * D_ + j3 + 1], iz = e[2 * D_ + j3 + 2];
      hx = rx / (1.f + __expf(-dx)) + ix;   // sigmoid(dx)*rx + ix
      hy = ry / (1.f + __expf(-dy)) + iy;
      hz = rz / (1.f + __expf(-dz)) + iz;
      sH[tt * HP + j3]     = hx;
      sH[tt * HP + j3 + 1] = hy;
      sH[tt * HP + j3 + 2] = hz;
    }
    __syncthreads();

    // OUT(16x4, padded to 16) = H(16x384) @ W_out(384x4).
    // Split-K across all 4 waves: 24 K=4 WMMAs each, two interleaved
    // accumulators to break the dependence chain; partials reduced in LDS.
    {
      v8f acc0 = {0.f, 0.f, 0.f, 0.f, 0.f, 0.f, 0.f, 0.f};
      v8f acc1 = {0.f, 0.f, 0.f, 0.f, 0.f, 0.f, 0.f, 0.f};
      const int kbeg = uwave * 24;           // 96 K-steps / 4 waves
#pragma unroll 4
      for (int kk = 0; kk < 24; kk += 2) {
        int kr0 = (kbeg + kk) * 4 + k0;
        int kr1 = (kbeg + kk + 1) * 4 + k0;
        v2f a2, b2;
        a2.x = sH[t16 * HP + kr0];
        a2.y = sH[t16 * HP + kr0 + 1];
        b2.x = (t16 < 4) ? sWout[kr0 * 4 + t16] : 0.f;
        b2.y = (t16 < 4) ? sWout[(kr0 + 1) * 4 + t16] : 0.f;
        acc0 = __builtin_amdgcn_wmma_f32_16x16x4_f32(
            false, a2, false, b2, (short)0, acc0, false, false);
        a2.x = sH[t16 * HP + kr1];
        a2.y = sH[t16 * HP + kr1 + 1];
        b2.x = (t16 < 4) ? sWout[kr1 * 4 + t16] : 0.f;
        b2.y = (t16 < 4) ? sWout[(kr1 + 1) * 4 + t16] : 0.f;
        acc1 = __builtin_amdgcn_wmma_f32_16x16x4_f32(
            false, a2, false, b2, (short)0, acc1, false, false);
      }
      acc0 = acc0 + acc1;
#pragma unroll
      for (int v = 0; v < 8; ++v)
        sPart[uwave * 256 + (v + mhi) * 16 + t16] = acc0[v];
    }
    __syncthreads();

    // Reduce the 4 wave partials, normalize over the 4 components, store.
    if (tid < 16) {
      float p[4];
#pragma unroll
      for (int c = 0; c < 4; ++c)
        p[c] = sPart[0 * 256 + tid * 16 + c] + sPart[1 * 256 + tid * 16 + c] +
               sPart[2 * 256 + tid * 16 + c] + sPart[3 * 256 + tid * 16 + c];
      float n = sqrtf(p[0] * p[0] + p[1] * p[1] + p[2] * p[2] + p[3] * p[3]);
      float rn = 1.f / fmaxf(n, EPSF);
      float4 o = {p[0] * rn, p[1] * rn, p[2] * rn, p[3] * rn};
      *(float4*)&out[((size_t)b * T_ + t0 + tid) * 4] = o;
    }
    __syncthreads();   // protect sE/sH/sPart before next chunk overwrites them
  }
}

extern "C" void kernel_launch(void* const* d_in, const int* in_sizes, int n_in,
                              void* d_out, int out_size, void* d_ws, size_t ws_size,
                              hipStream_t stream) {
  const float* quat  = (const float*)d_in[0];
  const float* W_in  = (const float*)d_in[1];
  const float* W_biv = (const float*)d_in[2];
  const float* b_biv = (const float*)d_in[3];
  const float* W_dec = (const float*)d_in[4];
  const float* b_dec = (const float*)d_in[5];
  const float* W_inj = (const float*)d_in[6];
  const float* b_inj = (const float*)d_in[7];
  const float* W_out = (const float*)d_in[8];
  float* out  = (float*)d_out;
  float* weff = (float*)d_ws;            // 4*1152 floats
  float* bias = weff + 4 * CPW;          // 1152 floats

  so3_prep_kernel<<<(4 * CPW + 255) / 256, 256, 0, stream>>>(
      W_in, W_biv, b_biv, W_dec, b_dec, W_inj, b_inj, weff, bias);
  so3_scan_kernel<<<B_, 128, 0, stream>>>(quat, weff, bias, W_out, out);
}